// LearnerCT_52871047413971
// MI455X (gfx1250) — compile-verified
//
#include <hip/hip_runtime.h>

// ============================================================================
// Lyapunov-candidate fused kernel for MI455X (gfx1250, wave32, WMMA).
//
// Reference math collapsed to 4 GEMMs + elementwise:
//   z1 = S @ W1^T + b1                     [N,256]  (K=64)
//   w  = Sdot @ W1^T                       [N,256]  (K=64)
//   z2 = (z1*z1) @ W2^T + b2               [N,256]  (K=256)
//   nn_out = sum_h W3[h] * z2[h]^2
//   t  = 2*W3 (.) z2
//   u  = t @ W2                            [N,256]  (K=256)
//   grad_nn . Sdot = sum_p u[p]*2*z1[p]*w[p]        (pure elementwise!)
//   F = sum S^2 ; SD = sum S.Sdot
//   V = nn_out*F ; Vdot = 2*nn_out*SD + F*(grad_nn.Sdot) ; circle = F
//
// fp32 everywhere (matches reference dtype; z^2 chains overflow f16), using
// V_WMMA_F32_16X16X4_F32. One wave = 16 rows; intermediates staged row-major
// in per-wave LDS so A-fragments are simple b64 reads.
// ============================================================================

typedef float v2f __attribute__((ext_vector_type(2)));
typedef float v4f __attribute__((ext_vector_type(4)));
typedef float v8f __attribute__((ext_vector_type(8)));

#define NROWS 16384
#define DD    64
#define HH    256

static __device__ __forceinline__ v8f wmma_f32(v2f a, v2f b, v8f c) {
  // 8-arg form: (neg_a, A, neg_b, B, c_mod, C, reuse_a, reuse_b)
  return __builtin_amdgcn_wmma_f32_16x16x4_f32(false, a, false, b,
                                               (short)0, c, false, false);
}

__global__ __launch_bounds__(32)
void lyapunov_fused(const float* __restrict__ S, const float* __restrict__ Sdot,
                    const float* __restrict__ W1, const float* __restrict__ b1,
                    const float* __restrict__ W2, const float* __restrict__ b2,
                    const float* __restrict__ W3, float* __restrict__ out)
{
  // Per-wave LDS staging (one wave per workgroup -> private).
  __shared__ float s_z1[16 * HH];      // z1, row-major [16][256]
  __shared__ float s_w [16 * HH];      // w = Sdot @ W1^T
  __shared__ float s_t [16 * HH];      // t = 2*W3 (.) z2
  __shared__ float s_red[2 * 16 * 16]; // nn_out / grad-dot partials
  __shared__ float s_fs [16 * 4];      // per-row {F,SD} half-sums

  const int lane = threadIdx.x;   // 0..31 (wave32)
  const int lo   = lane & 15;     // row (A/C/D "M=lane" half) / col within tile
  const int hi   = lane >> 4;     // K-pair select for A/B fragments
  const int row0 = blockIdx.x * 16;

  // ---- per-row reductions on raw inputs: F = sum S^2, SD = sum S.Sdot ----
  {
    const float* sr = S    + (size_t)(row0 + lo) * DD + 32 * hi;
    const float* dr = Sdot + (size_t)(row0 + lo) * DD + 32 * hi;
    float F = 0.f, SD = 0.f;
#pragma unroll
    for (int j = 0; j < 8; ++j) {
      v4f s4 = *(const v4f*)(sr + 4 * j);
      v4f d4 = *(const v4f*)(dr + 4 * j);
      F  += s4.x * s4.x + s4.y * s4.y + s4.z * s4.z + s4.w * s4.w;
      SD += s4.x * d4.x + s4.y * d4.y + s4.z * d4.z + s4.w * d4.w;
    }
    s_fs[lo * 4 + hi * 2 + 0] = F;
    s_fs[lo * 4 + hi * 2 + 1] = SD;
  }

  // ---- GEMM1a: z1 = S @ W1^T + b1   (16x256, K=64) -> LDS row-major ----
#pragma unroll 1
  for (int nt = 0; nt < 16; ++nt) {
    v8f c = {};
#pragma unroll
    for (int kb = 0; kb < DD / 4; ++kb) {
      const int k = kb * 4 + 2 * hi;
      v2f a = *(const v2f*)(S  + (size_t)(row0 + lo) * DD + k);      // A[m][k..k+1]
      v2f b = *(const v2f*)(W1 + (size_t)(nt * 16 + lo) * DD + k);   // B[k][n]=W1[n][k]
      c = wmma_f32(a, b, c);
    }
    const float bb = b1[nt * 16 + lo];
#pragma unroll
    for (int i = 0; i < 8; ++i)
      s_z1[(i + 8 * hi) * HH + nt * 16 + lo] = c[i] + bb;
  }

  // ---- GEMM1b: w = Sdot @ W1^T      (16x256, K=64) -> LDS row-major ----
#pragma unroll 1
  for (int nt = 0; nt < 16; ++nt) {
    v8f c = {};
#pragma unroll
    for (int kb = 0; kb < DD / 4; ++kb) {
      const int k = kb * 4 + 2 * hi;
      v2f a = *(const v2f*)(Sdot + (size_t)(row0 + lo) * DD + k);
      v2f b = *(const v2f*)(W1   + (size_t)(nt * 16 + lo) * DD + k);
      c = wmma_f32(a, b, c);
    }
#pragma unroll
    for (int i = 0; i < 8; ++i)
      s_w[(i + 8 * hi) * HH + nt * 16 + lo] = c[i];
  }

  float no_part[8];   // nn_out partials, indexed by m = i + 8*hi
  float gs_part[8];   // grad_nn . Sdot partials
#pragma unroll
  for (int i = 0; i < 8; ++i) { no_part[i] = 0.f; gs_part[i] = 0.f; }

  // ---- GEMM2: z2 = (z1^2) @ W2^T + b2 (K=256); nn_out += W3*z2^2; t -> LDS ----
#pragma unroll 1
  for (int nt = 0; nt < 16; ++nt) {
    v8f c = {};
#pragma unroll
    for (int kb = 0; kb < HH / 4; ++kb) {
      const int k = kb * 4 + 2 * hi;
      v2f a = *(const v2f*)&s_z1[lo * HH + k];     // square on the fly
      a.x *= a.x;
      a.y *= a.y;
      v2f b = *(const v2f*)(W2 + (size_t)(nt * 16 + lo) * HH + k); // W2[n][k..k+1]
      c = wmma_f32(a, b, c);
    }
    const float bb = b2[nt * 16 + lo];
    const float w3 = W3[nt * 16 + lo];
#pragma unroll
    for (int i = 0; i < 8; ++i) {
      const float z = c[i] + bb;
      no_part[i] += w3 * z * z;
      s_t[(i + 8 * hi) * HH + nt * 16 + lo] = 2.f * w3 * z;
    }
  }

  // ---- GEMM3: u = t @ W2 (K=256); gs += u * 2*z1 * w (elementwise, C-layout) ----
#pragma unroll 1
  for (int pt = 0; pt < 16; ++pt) {
    v8f c = {};
#pragma unroll
    for (int kb = 0; kb < HH / 4; ++kb) {
      const int k = kb * 4 + 2 * hi;
      v2f a = *(const v2f*)&s_t[lo * HH + k];
      v2f b;                                        // B[k][p] = W2[k][p] (strided)
      b.x = W2[(size_t)k       * HH + pt * 16 + lo];
      b.y = W2[(size_t)(k + 1) * HH + pt * 16 + lo];
      c = wmma_f32(a, b, c);
    }
#pragma unroll
    for (int i = 0; i < 8; ++i) {
      const int m = i + 8 * hi;
      const int p = pt * 16 + lo;
      const float v = c[i] * 2.f * s_z1[m * HH + p];
      gs_part[i] += v * s_w[m * HH + p];
    }
  }

  // ---- cross-lane reduction via LDS (intra-wave, DS ops are in-order) ----
#pragma unroll
  for (int i = 0; i < 8; ++i) {
    s_red[      (i + 8 * hi) * 16 + lo] = no_part[i];
    s_red[256 + (i + 8 * hi) * 16 + lo] = gs_part[i];
  }

  if (lane < 16) {
    const int m = lane;
    float no = 0.f, gs = 0.f;
#pragma unroll
    for (int j = 0; j < 16; ++j) {
      no += s_red[      m * 16 + j];
      gs += s_red[256 + m * 16 + j];
    }
    const float F  = s_fs[m * 4 + 0] + s_fs[m * 4 + 2];
    const float SD = s_fs[m * 4 + 1] + s_fs[m * 4 + 3];
    const int g = row0 + m;
    out[g]             = no * F;                     // V
    out[NROWS + g]     = 2.f * no * SD + F * gs;     // Vdot
    out[2 * NROWS + g] = F;                          // circle
  }
}

extern "C" void kernel_launch(void* const* d_in, const int* in_sizes, int n_in,
                              void* d_out, int out_size, void* d_ws, size_t ws_size,
                              hipStream_t stream) {
  (void)in_sizes; (void)n_in; (void)out_size; (void)d_ws; (void)ws_size;
  const float* S    = (const float*)d_in[0];
  const float* Sdot = (const float*)d_in[1];
  const float* W1   = (const float*)d_in[2];
  const float* b1   = (const float*)d_in[3];
  const float* W2   = (const float*)d_in[4];
  const float* b2   = (const float*)d_in[5];
  const float* W3   = (const float*)d_in[6];
  float* out = (float*)d_out;

  // One wave32 per workgroup owns a 16-row tile; 16384/16 = 1024 workgroups.
  lyapunov_fused<<<NROWS / 16, 32, 0, stream>>>(S, Sdot, W1, b1, W2, b2, W3, out);
}